// RGATConv_88897233092808
// MI455X (gfx1250) — compile-verified
//
#include <hip/hip_runtime.h>
#include <hip/hip_bf16.h>

// H == D == 16 in the reference
#define HD 16
#define NEG_SLOPE 0.2f

typedef __attribute__((ext_vector_type(2))) float v2f;
typedef __attribute__((ext_vector_type(8))) float v8f;

// Sign-aware float atomic max: positive floats compare correctly as int,
// negative floats compare reversed as uint. Init value must be -inf.
__device__ __forceinline__ void atomicMaxF32(float* addr, float val) {
    if (val >= 0.0f) {
        atomicMax((int*)addr, __float_as_int(val));
    } else {
        atomicMin((unsigned int*)addr, __float_as_uint(val));
    }
}

// ---------------------------------------------------------------------------
// Kernel 1: fold attention vectors into the typed weights.
//   vl[t][j] = sum_d attn_l[t][j][d]            (j indexes H, d indexes D)
//   Wl[t][k][h] = sum_j W[t][k][h*16+j] * vl[t][j]   -> [T, IN, 16]
// One block per edge type.
// ---------------------------------------------------------------------------
__global__ void __launch_bounds__(256)
rgat_prep(const float* __restrict__ W,
          const float* __restrict__ attn_l,
          const float* __restrict__ attn_r,
          float* __restrict__ Wl, float* __restrict__ Wr, int IN) {
    const int t = blockIdx.x;
    const int tid = threadIdx.x;
    __shared__ float svl[HD], svr[HD];
    if (tid < HD) {
        float a = 0.f;
        for (int d = 0; d < HD; ++d) a += attn_l[(t * HD + tid) * HD + d];
        svl[tid] = a;
    } else if (tid < 2 * HD) {
        const int j = tid - HD;
        float a = 0.f;
        for (int d = 0; d < HD; ++d) a += attn_r[(t * HD + j) * HD + d];
        svr[j] = a;
    }
    __syncthreads();
    const float* Wt = W + (size_t)t * IN * (HD * HD);
    float* Wlt = Wl + (size_t)t * IN * HD;
    float* Wrt = Wr + (size_t)t * IN * HD;
    for (int idx = tid; idx < IN * HD; idx += blockDim.x) {
        const int k = idx >> 4, h = idx & 15;
        const float* row = Wt + (size_t)k * (HD * HD) + h * HD;
        float al = 0.f, ar = 0.f;
#pragma unroll
        for (int j = 0; j < HD; ++j) { al += row[j] * svl[j]; ar += row[j] * svr[j]; }
        Wlt[k * HD + h] = al;
        Wrt[k * HD + h] = ar;
    }
}

// Init m = -inf, s = 0 (workspace is poisoned with 0xAA by the harness).
__global__ void rgat_init(float* __restrict__ m, float* __restrict__ s, int n) {
    const int i = blockIdx.x * blockDim.x + threadIdx.x;
    if (i < n) { m[i] = -__builtin_inff(); s[i] = 0.f; }
}

// ---------------------------------------------------------------------------
// Kernel 2: el|er = x @ Wl[etype] | Wr[etype] via V_WMMA_F32_16X16X4_F32.
// One wave32 per 16-row tile; two 16x16 f32 accumulators (el, er).
// A layout (16x4 f32): lanes 0-15 hold rows M=lane, VGPR0/1 = K0/K1;
//                      lanes 16-31 same rows, K2/K3.
// B layout (4x16 f32): lanes 0-15 hold cols N=lane, VGPR0/1 = K0/K1;
//                      lanes 16-31 K2/K3.  (mirror of A)
// etype is sorted, so mixed-type tiles are rare (<T) -> scalar fallback,
// taken wave-uniformly so EXEC is all-1s inside the WMMA path.
// ---------------------------------------------------------------------------
__global__ void __launch_bounds__(256)
rgat_gemm(const float* __restrict__ x, const int* __restrict__ etype,
          const float* __restrict__ Wl, const float* __restrict__ Wr,
          float* __restrict__ el, float* __restrict__ er, int N, int IN) {
    const int lane = threadIdx.x & 31;
    const int wave = threadIdx.x >> 5;
    const int tile = blockIdx.x * (blockDim.x >> 5) + wave;
    const int r0 = tile * 16;
    if (r0 >= N) return;                      // wave-uniform

    const int mrow = lane & 15;               // row within tile / col of B
    const int hi   = lane >> 4;               // lane-half selects K pair
    const bool full = (r0 + 16 <= N);
    const int t0 = etype[r0];
    bool uni = false;
    if (full) uni = __all(etype[r0 + mrow] == t0);   // wave-uniform result

    if (uni) {
        const float* Bl = Wl + (size_t)t0 * IN * HD;
        const float* Br = Wr + (size_t)t0 * IN * HD;
        const float* xr = x + (size_t)(r0 + mrow) * IN;
        v8f cl = {}; v8f cr = {};
        for (int k = 0; k < IN; k += 4) {
            const int kk = k + 2 * hi;
            v2f a = *(const v2f*)(xr + kk);                         // b64 load
            v2f bl = { Bl[(kk + 0) * HD + mrow], Bl[(kk + 1) * HD + mrow] };
            v2f br = { Br[(kk + 0) * HD + mrow], Br[(kk + 1) * HD + mrow] };
            cl = __builtin_amdgcn_wmma_f32_16x16x4_f32(
                     false, a, false, bl, (short)0, cl, false, false);
            cr = __builtin_amdgcn_wmma_f32_16x16x4_f32(
                     false, a, false, br, (short)0, cr, false, false);
        }
        // D layout: lane n=mrow, VGPR v -> row (v + 8*hi)
#pragma unroll
        for (int v = 0; v < 8; ++v) {
            const int row = r0 + v + 8 * hi;
            el[row * HD + mrow] = cl[v];
            er[row * HD + mrow] = cr[v];
        }
    } else {
        // Scalar fallback: lanes 0-15 compute el, lanes 16-31 compute er.
        const float* Wm  = (hi == 0) ? Wl : Wr;
        float*       outm = (hi == 0) ? el : er;
        for (int r = 0; r < 16; ++r) {
            const int row = r0 + r;
            if (row >= N) break;
            const int t = etype[row];
            const float* B  = Wm + (size_t)t * IN * HD;
            const float* xr = x + (size_t)row * IN;
            float acc = 0.f;
            for (int k = 0; k < IN; ++k) acc += xr[k] * B[k * HD + mrow];
            outm[row * HD + mrow] = acc;
        }
    }
}

// ---------------------------------------------------------------------------
// Edge passes. el/er/m/s are L2-resident (6.4 MB each); e is recomputed each
// pass instead of spilling 200+ MB to HBM. One thread per (edge, head).
// ---------------------------------------------------------------------------
__device__ __forceinline__ float edge_e(const int* src, const int* dst,
                                        const float* el, const float* er,
                                        int e, int h, int& di) {
    const int si = src[e];
    di = dst[e];
    float v = el[si * HD + h] + er[di * HD + h];
    return v > 0.f ? v : NEG_SLOPE * v;
}

__global__ void __launch_bounds__(256)
rgat_edge_max(const int* __restrict__ src, const int* __restrict__ dst,
              const float* __restrict__ el, const float* __restrict__ er,
              float* __restrict__ m, long long EH) {
    const long long idx = (long long)blockIdx.x * blockDim.x + threadIdx.x;
    if (idx >= EH) return;
    const int e = (int)(idx >> 4), h = (int)(idx & 15);
    int di;
    const float v = edge_e(src, dst, el, er, e, h, di);
    atomicMaxF32(&m[di * HD + h], v);
}

__global__ void __launch_bounds__(256)
rgat_edge_sum(const int* __restrict__ src, const int* __restrict__ dst,
              const float* __restrict__ el, const float* __restrict__ er,
              const float* __restrict__ m, float* __restrict__ s, long long EH) {
    const long long idx = (long long)blockIdx.x * blockDim.x + threadIdx.x;
    if (idx >= EH) return;
    const int e = (int)(idx >> 4), h = (int)(idx & 15);
    int di;
    const float v = edge_e(src, dst, el, er, e, h, di);
    atomicAdd(&s[di * HD + h], __expf(v - m[di * HD + h]));
}

__global__ void __launch_bounds__(256)
rgat_edge_out(const int* __restrict__ src, const int* __restrict__ dst,
              const float* __restrict__ el, const float* __restrict__ er,
              const float* __restrict__ m, const float* __restrict__ s,
              float* __restrict__ out, long long EH) {
    const long long idx = (long long)blockIdx.x * blockDim.x + threadIdx.x;
    if (idx >= EH) return;
    const int e = (int)(idx >> 4), h = (int)(idx & 15);
    int di;
    const float v = edge_e(src, dst, el, er, e, h, di);
    out[idx] = __expf(v - m[di * HD + h]) / s[di * HD + h];
}

// ---------------------------------------------------------------------------
extern "C" void kernel_launch(void* const* d_in, const int* in_sizes, int n_in,
                              void* d_out, int out_size, void* d_ws, size_t ws_size,
                              hipStream_t stream) {
    const float* x      = (const float*)d_in[0];
    const float* W      = (const float*)d_in[1];
    const float* attn_l = (const float*)d_in[2];
    const float* attn_r = (const float*)d_in[3];
    const int*   etype  = (const int*)d_in[4];
    const int*   src    = (const int*)d_in[5];
    const int*   dst    = (const int*)d_in[6];
    float* out = (float*)d_out;

    const int       N  = in_sizes[4];                 // 100000
    const long long E  = in_sizes[5];                 // 3200000
    const int       IN = in_sizes[0] / N;             // 256
    const int       T  = in_sizes[2] / (HD * HD);     // 8

    float* ws = (float*)d_ws;
    size_t off = 0;
    float* Wl   = ws + off; off += (size_t)T * IN * HD;
    float* Wr   = ws + off; off += (size_t)T * IN * HD;
    float* el   = ws + off; off += (size_t)N * HD;
    float* er   = ws + off; off += (size_t)N * HD;
    float* mbuf = ws + off; off += (size_t)N * HD;
    float* sbuf = ws + off; off += (size_t)N * HD;

    rgat_prep<<<T, 256, 0, stream>>>(W, attn_l, attn_r, Wl, Wr, IN);

    const int nh = N * HD;
    rgat_init<<<(nh + 255) / 256, 256, 0, stream>>>(mbuf, sbuf, nh);

    const int tiles = (N + 15) / 16;
    rgat_gemm<<<(tiles + 7) / 8, 256, 0, stream>>>(x, etype, Wl, Wr, el, er, N, IN);

    const long long EH = E * HD;
    const int blks = (int)((EH + 255) / 256);
    rgat_edge_max<<<blks, 256, 0, stream>>>(src, dst, el, er, mbuf, EH);
    rgat_edge_sum<<<blks, 256, 0, stream>>>(src, dst, el, er, mbuf, sbuf, EH);
    rgat_edge_out<<<blks, 256, 0, stream>>>(src, dst, el, er, mbuf, sbuf, out, EH);
}